// MaxUnpooling2D_25065429139638
// MI455X (gfx1250) — compile-verified
//
#include <hip/hip_runtime.h>

// MaxUnpooling2D for B=16, H=128, W=128, C=64, up=(2,2) -> out [16,256,256,64] fp32.
//
// Strategy: scatter-as-gather. Each thread owns one pooled cell (b,h,w) and a
// group of 4 consecutive channels. It loads its 4 updates + 4 mask indices once
// (b128, non-temporal) and writes all 4 output positions of its 2x2 window
// (4x b128 non-temporal stores), selecting per-channel between the update value
// and 0 by comparing the mask against each candidate flat index. Every output
// element is written exactly once -> no zero-fill pass, no atomics, single pass
// over ~384 MB => bandwidth-bound at HBM speed.

namespace {
constexpr int kB  = 16;
constexpr int kH  = 128;
constexpr int kW  = 128;
constexpr int kC  = 64;
constexpr int kUP = 2;
constexpr int kHO = kH * kUP;                       // 256
constexpr int kWO = kW * kUP;                       // 256
constexpr int kFlatOut = kHO * kWO * kC;            // 4,194,304 per batch
constexpr int kC4 = kC / 4;                         // 16 channel groups
constexpr int kBlock = 256;                         // 8 wave32 per block
constexpr unsigned kThreads = (unsigned)kB * kH * kW * kC4;  // 4,194,304
constexpr unsigned long long kInElems = (unsigned long long)kB * kH * kW * kC;
}  // namespace

typedef __attribute__((ext_vector_type(4))) float v4f;
typedef __attribute__((ext_vector_type(4))) int   v4i;

__global__ __launch_bounds__(kBlock)
void max_unpool2d_gather_kernel(const float* __restrict__ updates,
                                const int*   __restrict__ mask,
                                float*       __restrict__ out) {
  const unsigned tid = blockIdx.x * kBlock + threadIdx.x;

  // Decompose linear id: (((b*H + h)*W + w)*C4 + c4); all power-of-two shifts.
  const unsigned c0 = (tid & (kC4 - 1)) << 2;   // first channel of the group
  unsigned t = tid >> 4;                         // / kC4
  const unsigned w = t & (kW - 1); t >>= 7;
  const unsigned h = t & (kH - 1); t >>= 7;
  const unsigned b = t;                          // 0..15

  const unsigned long long in_off =
      (((unsigned long long)b * kH + h) * kW + w) * (unsigned)kC + c0;

  // Speculative streaming prefetch ~32 KB ahead (gfx1250 global_prefetch_b8).
  {
    const unsigned long long pf = in_off + (unsigned long long)(8 * kBlock * 4);
    if (pf < kInElems) {
      __builtin_prefetch(updates + pf, 0, 0);
      __builtin_prefetch(mask + pf, 0, 0);
    }
  }

  // Streamed once: non-temporal b128 loads.
  const v4f upd = __builtin_nontemporal_load((const v4f*)(updates + in_off));
  const v4i msk = __builtin_nontemporal_load((const v4i*)(mask + in_off));

  float* const outb = out + (unsigned long long)b * (unsigned)kFlatOut;
  const unsigned ho = h * kUP;
  const unsigned wo = w * kUP;

  // Write the full 2x2 window: each output element is produced exactly once.
#pragma unroll
  for (int dh = 0; dh < kUP; ++dh) {
#pragma unroll
    for (int dw = 0; dw < kUP; ++dw) {
      const int cand0 =
          (int)(((ho + (unsigned)dh) * (unsigned)kWO + (wo + (unsigned)dw)) *
                    (unsigned)kC + c0);
      v4f v;
#pragma unroll
      for (int j = 0; j < 4; ++j) {
        v[j] = (msk[j] == cand0 + j) ? upd[j] : 0.0f;
      }
      __builtin_nontemporal_store(v, (v4f*)(outb + (unsigned)cand0));
    }
  }
}

extern "C" void kernel_launch(void* const* d_in, const int* in_sizes, int n_in,
                              void* d_out, int out_size, void* d_ws, size_t ws_size,
                              hipStream_t stream) {
  (void)in_sizes; (void)n_in; (void)out_size; (void)d_ws; (void)ws_size;
  const float* updates = (const float*)d_in[0];
  const int*   mask    = (const int*)d_in[1];
  float*       out     = (float*)d_out;

  const dim3 grid(kThreads / kBlock);   // 16384 blocks
  const dim3 block(kBlock);             // 256 threads = 8 wave32
  max_unpool2d_gather_kernel<<<grid, block, 0, stream>>>(updates, mask, out);
}